// DecodePrediction_61237643706458
// MI455X (gfx1250) — compile-verified
//
#include <hip/hip_runtime.h>
#include <math.h>

// ---------------------------------------------------------------------------
// SSD decode + greedy NMS + top-k.
//
//  K2 NMS (1024 thr = 32 waves, 1 block/image):
//    - scores live in LDS (87.4 KB of the WGP's 320 KB); each element owned
//      by one thread -> in-order same-thread ds_load/ds_store, no barriers
//      needed around score access, bank-conflict-free (stride-1 lanes).
//      Init uses gfx1250 GLOBAL_LOAD_ASYNC_TO_LDS_B32 (ASYNCcnt-tracked
//      global->LDS DMA; builtin signature: (AS1 int*, AS3 int*, imm, cpol)).
//    - decoded boxes live in VGPRs: 4 x 22 = 88 regs (+~20 temps), no spills
//      (verified: clause-only prologue, no scratch ops).
//    - 100 serial rounds: ds argmax + wave32 shuffle + LDS tree reduction,
//      LDS broadcast of winner, division-free IoU suppression
//      (iou>t  <=>  uni>0 && inter > t*max(uni,1e-12)).
//    - 100 rounds == reference's 200 rounds + stable top_k(100): greedy-NMS
//      selected scores are non-increasing, so the first 100 rows ARE the
//      top-100. d_out is pre-zeroed with hipMemsetAsync, so early exit on
//      exhaustion is a bare uniform break.
//
//  K1 decode (256 thr): 1 thread/box. argmax over 21 logits, box regression,
//     writes SoA state to d_ws (float4 boxes / scores / class). Reads y_pred
//     once (~70 MB, HBM-floor ~3us) via b128 clauses.
// ---------------------------------------------------------------------------

namespace {
constexpr int   kNBoxes     = 21840;          // sum(fh*fw) * 4 aspect ratios
constexpr int   kRow        = 25;             // 21 logits + 4 box deltas
constexpr int   kClasses    = 21;
constexpr int   kDecThreads = 256;
constexpr int   kNmsThreads = 1024;           // 32 waves (wave32), == 1<<10
constexpr int   kItems      = (kNBoxes + kNmsThreads - 1) / kNmsThreads; // 22
constexpr int   kWaves      = kNmsThreads / 32;
constexpr int   kTopK       = 100;
constexpr float kConf       = 0.01f;
constexpr float kIouTh      = 0.45f;
constexpr float kNeg        = -1e30f;
constexpr float kXyVar      = 0.1f;
constexpr float kWhVar      = 0.2f;
}

#if defined(__AMDGCN__) && __has_builtin(__builtin_amdgcn_global_load_async_to_lds_b32)
#define NMS_ASYNC_LDS 1
typedef __attribute__((address_space(1))) int* as1_i32p;
typedef __attribute__((address_space(3))) int* as3_i32p;
#else
#define NMS_ASYNC_LDS 0
#endif

// ------------------------------- K2: NMS -----------------------------------
__global__ __launch_bounds__(kNmsThreads, 1)
void ssd_nms_kernel(const float4* __restrict__ ws_boxes,
                    const float*  __restrict__ ws_scores,
                    const float*  __restrict__ ws_cls,
                    float* __restrict__ out)
{
    __shared__ float sc[kNBoxes];            // 87,360 B of 320 KB WGP LDS
    __shared__ float red_v[kWaves];
    __shared__ int   red_i[kWaves];
    __shared__ float bc[8];                  // 0:val 1:x1 2:y1 3:x2 4:y2 5:area
    __shared__ int   bc_idx;

    const int img  = blockIdx.x;
    const int tid  = threadIdx.x & (kNmsThreads - 1);   // explicit [0,1023]
    const int lane = tid & 31;
    const int wave = tid >> 5;

    const float4* __restrict__ boxes  = ws_boxes  + (size_t)img * kNBoxes;
    const float*  __restrict__ scores = ws_scores + (size_t)img * kNBoxes;
    const float*  __restrict__ clsarr = ws_cls    + (size_t)img * kNBoxes;

    // Decoded boxes resident in VGPRs (4 x 22 = 88 regs).
    float bx1[kItems], by1[kItems], bx2[kItems], by2[kItems];

#if NMS_ASYNC_LDS
    // gfx1250 per-lane global->LDS DMA for the score table (ASYNCcnt).
    {
        as1_i32p gsc = (as1_i32p)(const_cast<float*>(scores));
        as3_i32p lsc = (as3_i32p)sc;
        #pragma unroll
        for (int k = 0; k < kItems; ++k) {
            const int j = tid + k * kNmsThreads;
            if (j < kNBoxes) {               // folds for k <= 20
                __builtin_amdgcn_global_load_async_to_lds_b32(gsc + j, lsc + j, 0, 0);
                const float4 b = boxes[j];   // coalesced global_load_b128
                bx1[k] = b.x; by1[k] = b.y; bx2[k] = b.z; by2[k] = b.w;
            } else {
                bx1[k] = 0.f; by1[k] = 0.f; bx2[k] = 0.f; by2[k] = 0.f;
            }
        }
    }
#if __has_builtin(__builtin_amdgcn_s_wait_asynccnt)
    __builtin_amdgcn_s_wait_asynccnt(0);
#else
    asm volatile("s_wait_asynccnt 0x0" ::: "memory");
#endif
#else
    #pragma unroll
    for (int k = 0; k < kItems; ++k) {
        const int j = tid + k * kNmsThreads;
        if (j < kNBoxes) {                   // folds for k <= 20
            const float4 b = boxes[j];       // coalesced global_load_b128
            bx1[k] = b.x; by1[k] = b.y; bx2[k] = b.z; by2[k] = b.w;
            sc[j]  = scores[j];              // global b32 -> ds_store_b32
        } else {
            bx1[k] = 0.f; by1[k] = 0.f; bx2[k] = 0.f; by2[k] = 0.f;
        }
    }
#endif
    __syncthreads();

    for (int iter = 0; iter < kTopK; ++iter) {
        // --- local argmax over owned LDS scores (lowest index wins ties) ---
        float lv = sc[tid]; int lk = 0;      // k = 0 always in range
        #pragma unroll
        for (int k = 1; k < kItems; ++k) {
            const int j = tid + k * kNmsThreads;
            if (j < kNBoxes) {
                const float v = sc[j];       // ds_load_b32, conflict-free
                if (v > lv) { lv = v; lk = k; }   // strict >: lowest k wins
            }
        }
        int li = tid | (lk << 10);           // global index, one OR

        // --- wave32 shuffle reduction (value first, then lowest index) ---
        #pragma unroll
        for (int off = 16; off > 0; off >>= 1) {
            const float ov = __shfl_down(lv, off, 32);
            const int   oi = __shfl_down(li, off, 32);
            if (ov > lv || (ov == lv && oi < li)) { lv = ov; li = oi; }
        }
        if (lane == 0) { red_v[wave] = lv; red_i[wave] = li; }
        __syncthreads();
        if (wave == 0) {                     // kWaves == 32: one entry/lane
            lv = red_v[lane];
            li = red_i[lane];
            #pragma unroll
            for (int off = 16; off > 0; off >>= 1) {
                const float ov = __shfl_down(lv, off, 32);
                const int   oi = __shfl_down(li, off, 32);
                if (ov > lv || (ov == lv && oi < li)) { lv = ov; li = oi; }
            }
            if (lane == 0) { bc[0] = lv; bc_idx = li; }
        }
        __syncthreads();

        const float bestVal = bc[0];
        const int   bestIdx = bc_idx;
        if (!(bestVal > kNeg * 0.5f)) break; // uniform; out is pre-zeroed

        const int  bestK = bestIdx >> 10;
        const bool mine  = (bestIdx & (kNmsThreads - 1)) == tid;

        // --- owner broadcasts the winning box ---
        if (mine) {
            bc[1] = bx1[bestK]; bc[2] = by1[bestK];
            bc[3] = bx2[bestK]; bc[4] = by2[bestK];
            bc[5] = (bx2[bestK] - bx1[bestK]) * (by2[bestK] - by1[bestK]);
        }
        __syncthreads();

        const float wx1 = bc[1], wy1 = bc[2], wx2 = bc[3], wy2 = bc[4];
        const float warea = bc[5];

        // --- division-free IoU suppression of owned elements ---
        #pragma unroll
        for (int k = 0; k < kItems; ++k) {
            const int j = tid + k * kNmsThreads;
            if (j < kNBoxes) {               // folds for k <= 20
                const float ix1 = fmaxf(wx1, bx1[k]);
                const float iy1 = fmaxf(wy1, by1[k]);
                const float ix2 = fminf(wx2, bx2[k]);
                const float iy2 = fminf(wy2, by2[k]);
                const float inter = fmaxf(ix2 - ix1, 0.f) * fmaxf(iy2 - iy1, 0.f);
                const float areaJ = (bx2[k] - bx1[k]) * (by2[k] - by1[k]);
                const float uni   = warea + areaJ - inter;
                // iou > t  <=>  uni > 0 && inter > t * max(uni, 1e-12)
                const bool  sup   = (uni > 0.f) && (inter > kIouTh * fmaxf(uni, 1e-12f));
                if (sup || (mine && k == bestK)) sc[j] = kNeg;  // ds_store_b32
            }
        }

        if (tid == 0) {
            float* o = out + ((size_t)img * kTopK + iter) * 6;
            o[0] = clsarr[bestIdx];   // class id (one L2-hot load)
            o[1] = bestVal;           // confidence
            o[2] = wx1; o[3] = wy1; o[4] = wx2; o[5] = wy2;
        }
        // Same-thread ds ordering makes the suppression stores visible to the
        // owner's own next-round ds_loads; cross-thread state (bc/red) is
        // fenced by the three __syncthreads above.
    }
}

// ------------------------------ K1: decode ---------------------------------
__global__ __launch_bounds__(kDecThreads)
void ssd_decode_kernel(const float* __restrict__ y_pred,
                       const float* __restrict__ anchors,
                       float4* __restrict__ ws_boxes,
                       float*  __restrict__ ws_scores,
                       float*  __restrict__ ws_cls)
{
    const int j   = blockIdx.x * kDecThreads + threadIdx.x;
    const int img = blockIdx.y;
    if (j >= kNBoxes) return;

    const float* __restrict__ r = y_pred + ((size_t)img * kNBoxes + j) * kRow;
    if (j + kDecThreads < kNBoxes) {
        // gfx1250: lowers to global_prefetch_b8
        __builtin_prefetch(r + (size_t)kDecThreads * kRow, 0, 3);
    }

    // argmax over logits (lowest index wins ties, matching jnp.argmax)
    float best = r[0]; int cls = 0;
    #pragma unroll
    for (int c = 1; c < kClasses; ++c) {
        const float v = r[c];
        if (v > best) { best = v; cls = c; }
    }

    const float tx = r[21], ty = r[22], tw = r[23], th = r[24];
    const float4 a = ((const float4* __restrict__)anchors)[j];   // cx,cy,w,h
    const float cx = tx * kXyVar * a.z + a.x;
    const float cy = ty * kXyVar * a.w + a.y;
    const float w  = expf(tw * kWhVar) * a.z;
    const float h  = expf(th * kWhVar) * a.w;

    float4 b;
    b.x = fminf(fmaxf(cx - 0.5f * w, 0.f), 1.f);
    b.y = fminf(fmaxf(cy - 0.5f * h, 0.f), 1.f);
    b.z = fminf(fmaxf(cx + 0.5f * w, 0.f), 1.f);
    b.w = fminf(fmaxf(cy + 0.5f * h, 0.f), 1.f);

    const size_t o = (size_t)img * kNBoxes + j;
    ws_boxes[o]  = b;
    ws_scores[o] = (cls != 0 && best > kConf) ? best : kNeg;
    ws_cls[o]    = (float)cls;
}

// ------------------------------ launcher -----------------------------------
extern "C" void kernel_launch(void* const* d_in, const int* in_sizes, int n_in,
                              void* d_out, int out_size, void* d_ws, size_t ws_size,
                              hipStream_t stream) {
    (void)n_in; (void)ws_size;
    const float* y_pred  = (const float*)d_in[0];
    const float* anchors = (const float*)d_in[1];
    float* out = (float*)d_out;

    const int batch = in_sizes[0] / (kNBoxes * kRow);   // 32

    // pre-zero output: rows past NMS exhaustion must be 0 (capturable async op)
    (void)hipMemsetAsync(out, 0, (size_t)out_size * sizeof(float), stream);

    // workspace layout (SoA): boxes | scores | cls
    float4* ws_boxes  = (float4*)d_ws;
    float*  ws_scores = (float*)((char*)d_ws + (size_t)batch * kNBoxes * sizeof(float4));
    float*  ws_cls    = ws_scores + (size_t)batch * kNBoxes;

    dim3 dgrid((kNBoxes + kDecThreads - 1) / kDecThreads, batch);
    ssd_decode_kernel<<<dgrid, kDecThreads, 0, stream>>>(
        y_pred, anchors, ws_boxes, ws_scores, ws_cls);

    ssd_nms_kernel<<<batch, kNmsThreads, 0, stream>>>(
        ws_boxes, ws_scores, ws_cls, out);
}